// Sampler_61323543053066
// MI455X (gfx1250) — compile-verified
//
#include <hip/hip_runtime.h>
#include <hip/hip_bf16.h>
#include <float.h>

// ---------------------------------------------------------------------------
// Gumbel-max sampler, rewritten as a single streaming pass:
//   argmax_i softmax(l/T)_i / E_i == argmax_i ( l_i/T - log(max(-log u_i,1e-10)) )
// Memory-bound: ~131 MB @ 23.3 TB/s => ~5.6 us floor. CDNA5 path used:
// async global->LDS staging (ASYNCcnt) + wave32 shuffle reduction.
// ---------------------------------------------------------------------------

#define B_ROWS 128
#define VOCAB 128000
#define SPLITS 25            // 128000 / 25 = 5120 floats per split
#define TPB 256              // 8 waves (wave32)
#define TILE 1024            // floats per tile per array = TPB * 4
#define TILES_PER_SPLIT 5    // 5 * 1024 = 5120

#if defined(__has_builtin)
#if __has_builtin(__builtin_amdgcn_global_load_async_to_lds_b128)
#define HAVE_ASYNC_B128 1
#endif
#if __has_builtin(__builtin_amdgcn_s_wait_asynccnt)
#define HAVE_WAIT_ASYNC 1
#endif
#endif
#ifndef HAVE_ASYNC_B128
#define HAVE_ASYNC_B128 0
#endif
#ifndef HAVE_WAIT_ASYNC
#define HAVE_WAIT_ASYNC 0
#endif

#if HAVE_ASYNC_B128
// Builtin signature (from clang diagnostic): params are pointers to 4 x i32
// vectors in AS(1) (global) and AS(3) (LDS).
typedef int v4i_t __attribute__((ext_vector_type(4)));
typedef __attribute__((address_space(1))) v4i_t* g_v4i_p;
typedef __attribute__((address_space(3))) v4i_t* l_v4i_p;
#endif

// Stage 16 bytes (one float4 per lane) from global into LDS.
__device__ __forceinline__ void stage16(const float* __restrict__ g, float* l) {
#if HAVE_ASYNC_B128
  __builtin_amdgcn_global_load_async_to_lds_b128((g_v4i_p)g, (l_v4i_p)l,
                                                 /*offset=*/0, /*cpol=*/0);
#else
  *(float4*)l = *(const float4*)g;   // synchronous fallback (VGPR bounce)
#endif
}

#if HAVE_WAIT_ASYNC
#define WAIT_ASYNC(n) __builtin_amdgcn_s_wait_asynccnt(n)
#elif HAVE_ASYNC_B128
#define WAIT_ASYNC(n) asm volatile("s_wait_asynccnt %0" ::"i"(n) : "memory")
#else
#define WAIT_ASYNC(n) do {} while (0)
#endif

__global__ __launch_bounds__(TPB) void sample_partial_kernel(
    const float* __restrict__ logits, const float* __restrict__ temps,
    const float* __restrict__ noise, float2* __restrict__ ws) {
  __shared__ float sL[2][TILE];
  __shared__ float sN[2][TILE];
  __shared__ float wv[TPB / 32];
  __shared__ int wi[TPB / 32];

  const int row = blockIdx.y;
  const int split = blockIdx.x;
  const int tid = threadIdx.x;

  const float invT = 1.0f / temps[row];
  const int split_base = split * (TILE * TILES_PER_SPLIT);
  const long long gbase = (long long)row * VOCAB + split_base;
  const int lbase = tid * 4;  // this lane's 16B slot in each tile

  const float* gL = logits + gbase + lbase;
  const float* gN = noise + gbase + lbase;

  // Prime the pipeline: tile 0 into buffer 0 (1 async op per array per wave).
  stage16(gL, &sL[0][lbase]);
  stage16(gN, &sN[0][lbase]);

  float bestv = -FLT_MAX;
  int besti = split_base + lbase;

#pragma unroll
  for (int t = 0; t < TILES_PER_SPLIT; ++t) {
    const int buf = t & 1;
    if (t + 1 < TILES_PER_SPLIT) {
      // Prefetch next tile into the other buffer, then wait until only the
      // newest tile's loads (2 per wave... 2 arrays) remain outstanding.
      stage16(gL + (t + 1) * TILE, &sL[(t + 1) & 1][lbase]);
      stage16(gN + (t + 1) * TILE, &sN[(t + 1) & 1][lbase]);
      WAIT_ASYNC(2);
    } else {
      WAIT_ASYNC(0);
    }
    // Each lane consumes exactly the 16B it staged: no cross-wave barrier.
    const float4 lv = *(const float4*)&sL[buf][lbase];
    const float4 nv = *(const float4*)&sN[buf][lbase];
    const int idx0 = split_base + t * TILE + lbase;

    const float kx = lv.x * invT - __logf(fmaxf(-__logf(nv.x), 1e-10f));
    const float ky = lv.y * invT - __logf(fmaxf(-__logf(nv.y), 1e-10f));
    const float kz = lv.z * invT - __logf(fmaxf(-__logf(nv.z), 1e-10f));
    const float kw = lv.w * invT - __logf(fmaxf(-__logf(nv.w), 1e-10f));

    if (kx > bestv) { bestv = kx; besti = idx0 + 0; }
    if (ky > bestv) { bestv = ky; besti = idx0 + 1; }
    if (kz > bestv) { bestv = kz; besti = idx0 + 2; }
    if (kw > bestv) { bestv = kw; besti = idx0 + 3; }
  }

  // Wave32 reduction; on ties prefer the lower index (argmax-first semantics).
#pragma unroll
  for (int off = 16; off > 0; off >>= 1) {
    const float ov = __shfl_xor(bestv, off, 32);
    const int oi = __shfl_xor(besti, off, 32);
    if (ov > bestv || (ov == bestv && oi < besti)) { bestv = ov; besti = oi; }
  }

  const int lane = tid & 31;
  const int wid = tid >> 5;
  if (lane == 0) { wv[wid] = bestv; wi[wid] = besti; }
  __syncthreads();
  if (tid == 0) {
#pragma unroll
    for (int w = 1; w < TPB / 32; ++w) {
      if (wv[w] > bestv || (wv[w] == bestv && wi[w] < besti)) {
        bestv = wv[w]; besti = wi[w];
      }
    }
    ws[row * SPLITS + split] = make_float2(bestv, __int_as_float(besti));
  }
}

__global__ __launch_bounds__(128) void sample_reduce_kernel(
    const float2* __restrict__ ws, float* __restrict__ out) {
  const int row = threadIdx.x;
  if (row >= B_ROWS) return;
  float bestv = -FLT_MAX;
  int besti = 0;
#pragma unroll
  for (int s = 0; s < SPLITS; ++s) {
    const float2 p = ws[row * SPLITS + s];
    if (p.x > bestv) { bestv = p.x; besti = __float_as_int(p.y); }
  }
  out[row] = (float)besti;  // indices < 2^24: exact in f32
}

extern "C" void kernel_launch(void* const* d_in, const int* in_sizes, int n_in,
                              void* d_out, int out_size, void* d_ws,
                              size_t ws_size, hipStream_t stream) {
  const float* logits = (const float*)d_in[0];
  const float* temps = (const float*)d_in[1];
  const float* noise = (const float*)d_in[2];
  float* out = (float*)d_out;
  float2* ws = (float2*)d_ws;  // SPLITS * B_ROWS * 8 bytes = 25.6 KB

  dim3 grid(SPLITS, B_ROWS, 1);
  sample_partial_kernel<<<grid, TPB, 0, stream>>>(logits, temps, noise, ws);
  sample_reduce_kernel<<<1, 128, 0, stream>>>(ws, out);
}